// postprocess_19739669692975
// MI455X (gfx1250) — compile-verified
//
#include <hip/hip_runtime.h>

// Postprocess for (B=32, C=85, N=25200) detector output.
// Live outputs (flat concat): bs=32 (int32 bits), int32 zero boxes [B,N,4],
// float32 score [B,N] with score>0.15 -> 0.
// Pure data movement: ~3.2 MB read + ~16.1 MB write => ~0.85 us @ 23.3 TB/s.
// No matrix contraction exists here, so WMMA is inapplicable; the MI455X-
// rewarded path is wide coalesced VMEM (b128) with 32-bit SADDR+voffset
// addressing and a single fused kernel (one graph node).

namespace {
constexpr unsigned kB = 32;
constexpr unsigned kC = 85;
constexpr unsigned kN = 25200;
constexpr float    kScoreThr = 0.15f;

constexpr unsigned kBoxElems   = kB * kN * 4u;        // 3,225,600 int32 zeros
constexpr unsigned kZeroRegion = 1u + kBoxElems;      // bs + boxes = 3,225,601
constexpr unsigned kScoreBase  = kZeroRegion;         // score starts here (4B-misaligned vs 16B)

// Zero region: 4 elements per thread via int4 stores.
constexpr unsigned kZeroThreads = (kZeroRegion + 3u) / 4u;        // 806,401

// Score region: pad 6300 quads/batch to 8192 so batch = s >> 13 (no division).
constexpr unsigned kQuadsPadded  = 8192;                          // >= 25200/4 = 6300
constexpr unsigned kScoreThreads = kB * kQuadsPadded;             // 262,144
constexpr unsigned kTotalThreads = kZeroThreads + kScoreThreads;  // 1,068,545
}

__global__ __launch_bounds__(256)
void postprocess_fused_kernel(const float* __restrict__ in, int* __restrict__ out) {
    unsigned t = blockIdx.x * 256u + threadIdx.x;

    if (t < kZeroThreads) {
        // --- bs + boxes: 16B-aligned int4 zero stores (global_store_b128) ---
        unsigned base = t * 4u;
        if (base + 3u < kZeroRegion) {
            int4 z;
            z.x = (base == 0u) ? (int)kB : 0;   // raw int32 bits of bs=32 at element 0
            z.y = 0; z.z = 0; z.w = 0;
            *reinterpret_cast<int4*>(out + base) = z;
        } else if (base < kZeroRegion) {
            // scalar tail: single element (index 3,225,600)
            for (unsigned j = base; j < kZeroRegion; ++j) out[j] = 0;
        }
    } else {
        // --- score: float4 load of output[b, 4, n4..n4+3], thresholded ---
        unsigned s  = t - kZeroThreads;
        unsigned b  = s >> 13;              // s / 8192
        unsigned n4 = (s & 8191u) * 4u;     // padded; discard past real row
        if (n4 >= kN) return;

        // source element offset b*C*N + 4*N + n4 is 16B-aligned for all b, n4
        const float4 v = *reinterpret_cast<const float4*>(
            in + b * (kC * kN) + 4u * kN + n4);          // global_load_b128

        // dest region starts at odd-DWORD offset => four contiguous b32 stores
        float* dst = reinterpret_cast<float*>(out) + kScoreBase + b * kN + n4;
        dst[0] = (v.x > kScoreThr) ? 0.0f : v.x;
        dst[1] = (v.y > kScoreThr) ? 0.0f : v.y;
        dst[2] = (v.z > kScoreThr) ? 0.0f : v.z;
        dst[3] = (v.w > kScoreThr) ? 0.0f : v.w;
    }
}

extern "C" void kernel_launch(void* const* d_in, const int* in_sizes, int n_in,
                              void* d_out, int out_size, void* d_ws, size_t ws_size,
                              hipStream_t stream) {
    (void)in_sizes; (void)n_in; (void)out_size; (void)d_ws; (void)ws_size;
    const float* in = (const float*)d_in[0];
    int* out = (int*)d_out;

    const unsigned block = 256;                                   // 8 wave32 waves
    const unsigned grid  = (kTotalThreads + block - 1u) / block;  // 4,175 blocks
    postprocess_fused_kernel<<<dim3(grid), dim3(block), 0, stream>>>(in, out);
}